// Mamba2PathReader_51840255263257
// MI455X (gfx1250) — compile-verified
//
#include <hip/hip_runtime.h>
#include <hip/hip_bf16.h>

typedef __attribute__((ext_vector_type(2))) float v2f;
typedef __attribute__((ext_vector_type(8))) float v8f;

#define BB 8
#define TT 2048
#define DIM 512
#define DM 128
#define NSTATE 16
#define HDIM 64
#define NH 4
#define DINNER 256
#define CONVDIM 288
#define PROJ 548
#define ROWS (BB * TT)   // 16384
#define EPS 1e-5f

// ---------------------------------------------------------------------------
// FP32 WMMA GEMM: Out[M,N] = A[M,K] @ W[K,N] (+bias) (+resid)
// Block = 128 threads = 4 waves; each wave computes a 32x32 tile.
// Uses V_WMMA_F32_16X16X4_F32 (full fp32 precision matrix pipe).
// N, K are compile-time so all B-fragment strides fold into the 24-bit
// immediate offsets of global_load -> inner loop is pure load+wmma.
// Edge columns (N=548 case) are clamped before the loop (a B-fragment lane
// only feeds its own output column) and masked at store time.
// ---------------------------------------------------------------------------
template <int N>
__device__ inline void store_tile(v8f acc, int row0, int col,
                                  const float* __restrict__ bias,
                                  const float* __restrict__ resid,
                                  float* __restrict__ Out) {
    if (col >= N) return;
    float bv = bias ? bias[col] : 0.0f;
#pragma unroll
    for (int r = 0; r < 8; ++r) {
        size_t idx = (size_t)(row0 + r) * N + col;
        float v = acc[r] + bv;
        if (resid) v += resid[idx];
        Out[idx] = v;
    }
}

template <int N, int K>
__global__ __launch_bounds__(128)
void gemm_f32_wmma(const float* __restrict__ A, const float* __restrict__ W,
                   const float* __restrict__ bias, const float* __restrict__ resid,
                   float* __restrict__ Out, int M) {
    static_assert((K & 15) == 0, "K must be a multiple of 16");
    const int lane = threadIdx.x & 31;
    const int wave = threadIdx.x >> 5;
    const int wm = wave >> 1, wn = wave & 1;
    const int mBase = blockIdx.y * 64 + wm * 32;
    const int nBase = blockIdx.x * 64 + wn * 32;
    if (mBase >= M) return;

    const int r15 = lane & 15;
    const int khalf = (lane >> 4) << 1;   // 0 or 2 (K-pair select per lane half)

    // A fragment pointers (two 16-row slabs), pre-offset by khalf
    const float* a0p = A + (size_t)(mBase + r15) * K + khalf;
    const float* a1p = A + (size_t)(mBase + 16 + r15) * K + khalf;

    // Edge columns clamped into range: branch-free loads, masked at store.
    int c0 = nBase + r15;        if (c0 > N - 1) c0 = N - 1;
    int c1 = nBase + 16 + r15;   if (c1 > N - 1) c1 = N - 1;
    const float* w0p = W + (size_t)khalf * N + c0;
    const float* w1p = W + (size_t)khalf * N + c1;

    v8f acc00 = {}, acc01 = {}, acc10 = {}, acc11 = {};

    for (int k0 = 0; k0 < K; k0 += 16) {
#pragma unroll
        for (int u = 0; u < 4; ++u) {
            const int k = u * 4;                 // compile-time constant offsets
            v2f a0, a1, b0, b1;
            a0.x = a0p[k];  a0.y = a0p[k + 1];
            a1.x = a1p[k];  a1.y = a1p[k + 1];
            b0.x = w0p[k * N];  b0.y = w0p[(k + 1) * N];
            b1.x = w1p[k * N];  b1.y = w1p[(k + 1) * N];
            acc00 = __builtin_amdgcn_wmma_f32_16x16x4_f32(false, a0, false, b0, (short)0, acc00, false, false);
            acc01 = __builtin_amdgcn_wmma_f32_16x16x4_f32(false, a0, false, b1, (short)0, acc01, false, false);
            acc10 = __builtin_amdgcn_wmma_f32_16x16x4_f32(false, a1, false, b0, (short)0, acc10, false, false);
            acc11 = __builtin_amdgcn_wmma_f32_16x16x4_f32(false, a1, false, b1, (short)0, acc11, false, false);
        }
        a0p += 16;  a1p += 16;
        w0p += (size_t)16 * N;  w1p += (size_t)16 * N;
    }

    const int rOff = 8 * (lane >> 4);
    store_tile<N>(acc00, mBase + rOff,      nBase + r15,      bias, resid, Out);
    store_tile<N>(acc01, mBase + rOff,      nBase + 16 + r15, bias, resid, Out);
    store_tile<N>(acc10, mBase + 16 + rOff, nBase + r15,      bias, resid, Out);
    store_tile<N>(acc11, mBase + 16 + rOff, nBase + 16 + r15, bias, resid, Out);
}

// ---------------------------------------------------------------------------
// RMSNorm over last dim 128: one wave32 per row, 4 elements/lane.
// ---------------------------------------------------------------------------
__global__ __launch_bounds__(256)
void rmsnorm128(const float* __restrict__ x, const float* __restrict__ w,
                float* __restrict__ out, int rows) {
    const int lane = threadIdx.x & 31;
    const int row = blockIdx.x * 8 + (threadIdx.x >> 5);
    if (row >= rows) return;
    const float* xr = x + (size_t)row * DM;
    float v[4];
    float ss = 0.0f;
#pragma unroll
    for (int i = 0; i < 4; ++i) {
        v[i] = xr[lane + 32 * i];
        ss += v[i] * v[i];
    }
#pragma unroll
    for (int off = 16; off > 0; off >>= 1) ss += __shfl_xor(ss, off, 32);
    const float inv = rsqrtf(ss * (1.0f / DM) + EPS);
    float* o = out + (size_t)row * DM;
#pragma unroll
    for (int i = 0; i < 4; ++i) {
        int c = lane + 32 * i;
        o[c] = v[i] * inv * w[c];
    }
}

// ---------------------------------------------------------------------------
// Depthwise causal conv (K=4) over hBC channels + bias + SiLU.
// hBC lives in proj[:, 256:544] (row stride PROJ).
// ---------------------------------------------------------------------------
__global__ __launch_bounds__(256)
void conv_silu(const float* __restrict__ proj, const float* __restrict__ cw,
               const float* __restrict__ cb, float* __restrict__ out) {
    const int idx = blockIdx.x * blockDim.x + threadIdx.x;
    if (idx >= ROWS * CONVDIM) return;
    const int c = idx % CONVDIM;
    const int t = (idx / CONVDIM) % TT;
    const int b = idx / (CONVDIM * TT);
    float acc = cb[c];
#pragma unroll
    for (int k = 0; k < 4; ++k) {
        const int tt = t - 3 + k;
        if (tt >= 0)
            acc += proj[((size_t)(b * TT + tt)) * PROJ + DINNER + c] * cw[c * 4 + k];
    }
    out[idx] = acc / (1.0f + __expf(-acc));   // SiLU
}

// ---------------------------------------------------------------------------
// Sequential SSD scan: one block per (batch, head); 256 threads.
// Thread owns p = tid&63, 4 states (n = 4*(tid>>6)..+3) in registers.
// ---------------------------------------------------------------------------
__global__ __launch_bounds__(256)
void ssd_scan(const float* __restrict__ hbc, const float* __restrict__ proj,
              const float* __restrict__ A_log, const float* __restrict__ dt_bias,
              const float* __restrict__ Dh, float* __restrict__ y) {
    const int b = blockIdx.x / NH;
    const int h = blockIdx.x % NH;
    const int tid = threadIdx.x;
    const int p = tid & 63;
    const int ng = tid >> 6;

    const float Av  = -__expf(A_log[h]);
    const float dtb = dt_bias[h];
    const float Dv  = Dh[h];

    float st[4] = {0.0f, 0.0f, 0.0f, 0.0f};
    __shared__ float red[256];

    const float* hb = hbc  + (size_t)b * TT * CONVDIM;
    const float* pb = proj + (size_t)b * TT * PROJ;
    float* yo       = y    + (size_t)b * TT * DINNER + h * HDIM;

    for (int t = 0; t < TT; ++t) {
        const float* row = hb + (size_t)t * CONVDIM;
        float dtraw = pb[(size_t)t * PROJ + DINNER + CONVDIM + h] + dtb;
        float dtv = (dtraw > 20.0f) ? dtraw : log1pf(__expf(dtraw));
        float dAv = __expf(dtv * Av);
        float xv  = row[h * HDIM + p];
        float partial = 0.0f;
#pragma unroll
        for (int j = 0; j < 4; ++j) {
            const int n = ng * 4 + j;
            const float Bv = row[DINNER + n];            // B state coeffs
            const float Cv = row[DINNER + NSTATE + n];   // C readout coeffs
            st[j] = st[j] * dAv + dtv * xv * Bv;
            partial += st[j] * Cv;
        }
        red[tid] = partial;
        __syncthreads();
        if (ng == 0) {
            yo[(size_t)t * DINNER + p] =
                red[p] + red[p + 64] + red[p + 128] + red[p + 192] + xv * Dv;
        }
        __syncthreads();
    }
}

// ---------------------------------------------------------------------------
// Gated RMSNorm over DINNER=256: y = rmsnorm(y * silu(gate)) * w  (in place ok)
// ---------------------------------------------------------------------------
__global__ __launch_bounds__(256)
void gated_rmsnorm256(float* __restrict__ y, const float* __restrict__ proj,
                      const float* __restrict__ w, int rows) {
    const int lane = threadIdx.x & 31;
    const int row = blockIdx.x * 8 + (threadIdx.x >> 5);
    if (row >= rows) return;
    float* yr = y + (size_t)row * DINNER;
    const float* gr = proj + (size_t)row * PROJ;   // gate = cols [0,256)
    float v[8];
    float ss = 0.0f;
#pragma unroll
    for (int i = 0; i < 8; ++i) {
        const int c = lane + 32 * i;
        const float g = gr[c];
        const float t = yr[c] * (g / (1.0f + __expf(-g)));
        v[i] = t;
        ss += t * t;
    }
#pragma unroll
    for (int off = 16; off > 0; off >>= 1) ss += __shfl_xor(ss, off, 32);
    const float inv = rsqrtf(ss * (1.0f / DINNER) + EPS);
#pragma unroll
    for (int i = 0; i < 8; ++i) {
        const int c = lane + 32 * i;
        yr[c] = v[i] * inv * w[c];
    }
}

// ---------------------------------------------------------------------------
// pooled[b,d] = mean_t hn[b,t,d]
// ---------------------------------------------------------------------------
__global__ __launch_bounds__(256)
void pool_mean(const float* __restrict__ hn, float* __restrict__ pooled) {
    const int idx = blockIdx.x * blockDim.x + threadIdx.x;
    if (idx >= BB * DM) return;
    const int d = idx % DM;
    const int b = idx / DM;
    const float* p = hn + (size_t)b * TT * DM + d;
    float s = 0.0f;
    for (int t = 0; t < TT; ++t) s += p[(size_t)t * DM];
    pooled[idx] = s * (1.0f / TT);
}

// ---------------------------------------------------------------------------
// out[b,j] = pooled[b,:] @ out_w[:,j] + out_b[j]   (8x512, tiny)
// ---------------------------------------------------------------------------
__global__ __launch_bounds__(256)
void final_proj(const float* __restrict__ pooled, const float* __restrict__ ow,
                const float* __restrict__ ob, float* __restrict__ out) {
    const int idx = blockIdx.x * blockDim.x + threadIdx.x;
    if (idx >= BB * DIM) return;
    const int j = idx % DIM;
    const int b = idx / DIM;
    float s = ob[j];
#pragma unroll 4
    for (int d = 0; d < DM; ++d) s += pooled[b * DM + d] * ow[d * DIM + j];
    out[idx] = s;
}

// ---------------------------------------------------------------------------

extern "C" void kernel_launch(void* const* d_in, const int* in_sizes, int n_in,
                              void* d_out, int out_size, void* d_ws, size_t ws_size,
                              hipStream_t stream) {
    const float* x        = (const float*)d_in[0];
    const float* in_w     = (const float*)d_in[1];
    const float* in_b     = (const float*)d_in[2];
    const float* norm_w   = (const float*)d_in[3];
    const float* inproj_w = (const float*)d_in[4];
    const float* conv_w   = (const float*)d_in[5];
    const float* conv_b   = (const float*)d_in[6];
    const float* A_log    = (const float*)d_in[7];
    const float* Dp       = (const float*)d_in[8];
    const float* dt_bias  = (const float*)d_in[9];
    const float* gnorm_w  = (const float*)d_in[10];
    const float* outproj_w= (const float*)d_in[11];
    const float* normf_w  = (const float*)d_in[12];
    const float* out_w    = (const float*)d_in[13];
    const float* out_b    = (const float*)d_in[14];
    float* out = (float*)d_out;

    float* ws = (float*)d_ws;
    float* h      = ws;                               // ROWS*DM
    float* hn     = h    + (size_t)ROWS * DM;         // ROWS*DM
    float* proj   = hn   + (size_t)ROWS * DM;         // ROWS*PROJ
    float* hbc    = proj + (size_t)ROWS * PROJ;       // ROWS*CONVDIM
    float* yb     = hbc  + (size_t)ROWS * CONVDIM;    // ROWS*DINNER
    float* pooled = yb   + (size_t)ROWS * DINNER;     // BB*DM

    const dim3 blk128(128);
    const dim3 blk256(256);

    // h = x @ in_w + in_b   [16384,512]x[512,128]
    gemm_f32_wmma<DM, DIM><<<dim3(DM / 64, ROWS / 64), blk128, 0, stream>>>(
        x, in_w, in_b, nullptr, h, ROWS);

    for (int l = 0; l < 2; ++l) {
        rmsnorm128<<<ROWS / 8, blk256, 0, stream>>>(h, norm_w + l * DM, hn, ROWS);

        // proj = hn @ inproj_w[l]   [16384,128]x[128,548]
        gemm_f32_wmma<PROJ, DM><<<dim3((PROJ + 63) / 64, ROWS / 64), blk128, 0, stream>>>(
            hn, inproj_w + (size_t)l * DM * PROJ, nullptr, nullptr, proj, ROWS);

        conv_silu<<<(ROWS * CONVDIM + 255) / 256, blk256, 0, stream>>>(
            proj, conv_w + (size_t)l * CONVDIM * 4, conv_b + l * CONVDIM, hbc);

        ssd_scan<<<BB * NH, blk256, 0, stream>>>(
            hbc, proj, A_log + l * NH, dt_bias + l * NH, Dp + l * NH, yb);

        gated_rmsnorm256<<<ROWS / 8, blk256, 0, stream>>>(
            yb, proj, gnorm_w + l * DINNER, ROWS);

        // h = h + yb @ outproj_w[l]   [16384,256]x[256,128]
        gemm_f32_wmma<DM, DINNER><<<dim3(DM / 64, ROWS / 64), blk128, 0, stream>>>(
            yb, outproj_w + (size_t)l * DINNER * DM, nullptr, h, h, ROWS);
    }

    rmsnorm128<<<ROWS / 8, blk256, 0, stream>>>(h, normf_w, hn, ROWS);
    pool_mean<<<(BB * DM + 255) / 256, blk256, 0, stream>>>(hn, pooled);
    final_proj<<<(BB * DIM + 255) / 256, blk256, 0, stream>>>(pooled, out_w, out_b, out);
}